// BoxCrossCategoryLoss_25400436588780
// MI455X (gfx1250) — compile-verified
//
#include <hip/hip_runtime.h>
#include <hip/hip_bf16.h>

#define TPB 256

typedef __attribute__((ext_vector_type(2))) float v2f;
typedef __attribute__((ext_vector_type(8))) float v8f;

// ---------------- recipe tables -> 512-entry LUT on key = (cx<<6)|(cy<<3)|cz ---
// byte encoding: 0x80 = member of LOSS_RECIPE, 0x40 = member of NEG_LOSS_RECIPE,
// low 6 bits = NEG triple id.
struct Lut { unsigned char v[512]; };

constexpr int POS_R[14][3] = {
  {0,4,4},{0,6,4},{1,5,5},{1,6,5},{2,4,4},{2,5,5},{2,6,6},{2,7,7},
  {4,0,4},{4,2,4},{5,1,5},{5,2,5},{6,2,6},{7,2,7}};
constexpr int NEG_R[22][3] = {
  {0,4,1},{0,4,2},{0,6,1},{0,6,2},{1,5,0},{1,5,2},{1,6,0},{1,6,2},
  {2,4,1},{2,4,2},{2,5,0},{2,5,2},{4,0,1},{4,0,2},{4,2,1},{4,2,2},
  {5,1,0},{5,1,2},{5,2,0},{5,2,2},{2,7,2},{7,2,2}};

constexpr Lut makeLut() {
  Lut t{};
  for (int i = 0; i < 512; ++i) t.v[i] = 0;
  for (int j = 0; j < 14; ++j) {
    int k = (POS_R[j][0] << 6) | (POS_R[j][1] << 3) | POS_R[j][2];
    t.v[k] = (unsigned char)(t.v[k] | 0x80);
  }
  for (int j = 0; j < 22; ++j) {
    int k = (NEG_R[j][0] << 6) | (NEG_R[j][1] << 3) | NEG_R[j][2];
    t.v[k] = (unsigned char)(t.v[k] | 0x40 | j);
  }
  return t;
}

__device__ __constant__ Lut g_lut = makeLut();
// f-indices per NEG triple (f1 = xy>>2, f2 = yz>>2, f3 = xz>>2)
__device__ __constant__ int g_negf[22][3] = {
  {0,1,0},{0,1,0},{0,1,0},{0,1,0},{0,1,0},{0,1,0},{0,1,0},{0,1,0},
  {0,1,0},{0,1,0},{0,1,0},{0,1,0},{1,0,0},{1,0,0},{1,0,0},{1,0,0},
  {1,0,0},{1,0,0},{1,0,0},{1,0,0},{0,1,0},{1,0,0}};

struct NegRec {
  unsigned count;
  unsigned pad;
  unsigned long long mins;  // packed {min2:u32, min1:u32}, init all-ones
};

// cls: (r0,r1) -> (1,0):0 (0,1):1 (1,1):2 (0,0):3 ; nibble LUT 0x2013
__device__ __forceinline__ int clsOf(int r0, int r1) {
  return (0x2013 >> (((r0 << 1) | r1) << 2)) & 0xF;
}

__device__ __forceinline__ float log1mexpf_dev(float x) {
  // LOG_HALF = -0.6931471805599453
  return (x > -0.69314718f) ? __logf(-expm1f(x)) : log1pf(-__expf(x));
}

__global__ void init_ws(NegRec* neg) {
  int t = threadIdx.x;
  if (t < 22) {
    neg[t].count = 0u;
    neg[t].pad   = 0u;
    neg[t].mins  = 0xFFFFFFFFFFFFFFFFull;
  }
}

__device__ __forceinline__ void process_elem(
    int idx, int xy0, int xy1, int yz0, int yz1, int xz0, int xz1, int flag,
    const float* __restrict__ v1, const float* __restrict__ v2,
    const float* __restrict__ v3, float& acc, NegRec* neg) {
  int fb = flag << 2;
  int cx = clsOf(xy0, xy1) | fb;
  int cy = clsOf(yz0, yz1) | fb;
  int cz = clsOf(xz0, xz1) | fb;
  int key = (cx << 6) | (cy << 3) | cz;
  unsigned char b = g_lut.v[key];
  if (b & 0x80) {  // LOSS_RECIPE member (provably never with shared flag)
    int f1 = cx >> 2, f2 = cy >> 2, f3 = cz >> 2;
    acc += v1[2 * idx + f1] + v2[2 * idx + f2] - v3[2 * idx + f3];
  }
  if (b & 0x40) {  // NEG_LOSS_RECIPE member (provably never with shared flag)
    int j = b & 0x3F;
    atomicAdd(&neg[j].count, 1u);
    unsigned long long* p = &neg[j].mins;
    unsigned long long old = *(volatile unsigned long long*)p, assumed;
    unsigned u = (unsigned)idx;
    do {
      assumed = old;
      unsigned m1 = (unsigned)assumed, m2 = (unsigned)(assumed >> 32);
      unsigned n1, n2;
      if (u < m1)      { n1 = u;  n2 = m1; }
      else if (u < m2) { n1 = m1; n2 = u;  }
      else break;
      old = atomicCAS(p, assumed, ((unsigned long long)n2 << 32) | n1);
    } while (old != assumed);
  }
}

__global__ void __launch_bounds__(TPB)
loss_main(const float* __restrict__ v1, const float* __restrict__ v2,
          const float* __restrict__ v3, const int* __restrict__ xy,
          const int* __restrict__ yz, const int* __restrict__ xz,
          const int* __restrict__ flg, int B, int B4,
          float* __restrict__ partials, NegRec* neg) {
  float acc = 0.0f;
  int nthreads = gridDim.x * blockDim.x;
  for (int g = blockIdx.x * blockDim.x + threadIdx.x; g < B4; g += nthreads) {
    int base = 4 * g;
    if (base + 3 < B) {
      // 128-bit streaming loads: 28 B per element, memory-bound hot path
      int4 f4  = ((const int4*)flg)[g];
      int4 xyA = ((const int4*)xy)[2 * g], xyB = ((const int4*)xy)[2 * g + 1];
      int4 yzA = ((const int4*)yz)[2 * g], yzB = ((const int4*)yz)[2 * g + 1];
      int4 xzA = ((const int4*)xz)[2 * g], xzB = ((const int4*)xz)[2 * g + 1];
      process_elem(base + 0, xyA.x, xyA.y, yzA.x, yzA.y, xzA.x, xzA.y, f4.x, v1, v2, v3, acc, neg);
      process_elem(base + 1, xyA.z, xyA.w, yzA.z, yzA.w, xzA.z, xzA.w, f4.y, v1, v2, v3, acc, neg);
      process_elem(base + 2, xyB.x, xyB.y, yzB.x, yzB.y, xzB.x, xzB.y, f4.z, v1, v2, v3, acc, neg);
      process_elem(base + 3, xyB.z, xyB.w, yzB.z, yzB.w, xzB.z, xzB.w, f4.w, v1, v2, v3, acc, neg);
    } else {
      for (int e = base; e < B; ++e)
        process_elem(e, xy[2*e], xy[2*e+1], yz[2*e], yz[2*e+1], xz[2*e], xz[2*e+1],
                     flg[e], v1, v2, v3, acc, neg);
    }
  }

  // ---- wave32 reduction via V_WMMA_F32_16X16X4_F32 -------------------------
  // A holds one partial per lane (second K-slot zeroed), B = all-ones.
  // D[m,n] = row-sum of A, so summing a lane's 8 D registers yields the sum of
  // one 16-lane half; one xor-16 permute completes the 32-lane wave sum.
  float half_sum;
#if defined(__gfx1250__) && __has_builtin(__builtin_amdgcn_wmma_f32_16x16x4_f32)
  {
    v2f a; a[0] = acc; a[1] = 0.0f;
    v2f bOnes; bOnes[0] = 1.0f; bOnes[1] = 1.0f;
    v8f c = {};
    c = __builtin_amdgcn_wmma_f32_16x16x4_f32(
        /*neg_a=*/false, a, /*neg_b=*/false, bOnes,
        /*c_mod=*/(short)0, c, /*reuse_a=*/false, /*reuse_b=*/false);
    half_sum = ((c[0] + c[1]) + (c[2] + c[3])) + ((c[4] + c[5]) + (c[6] + c[7]));
  }
#else
  half_sum = acc;
  half_sum += __shfl_xor(half_sum, 1, 32);
  half_sum += __shfl_xor(half_sum, 2, 32);
  half_sum += __shfl_xor(half_sum, 4, 32);
  half_sum += __shfl_xor(half_sum, 8, 32);
#endif
  float wave_sum = half_sum + __shfl_xor(half_sum, 16, 32);

  __shared__ float wsums[TPB / 32];
  int wid = threadIdx.x >> 5, lid = threadIdx.x & 31;
  if (lid == 0) wsums[wid] = wave_sum;
  __syncthreads();
  if (threadIdx.x == 0) {
    float s = 0.0f;
    for (int w = 0; w < TPB / 32; ++w) s += wsums[w];
    partials[blockIdx.x] = s;
  }
}

__global__ void __launch_bounds__(TPB)
loss_final(const float* __restrict__ v1, const float* __restrict__ v2,
           const float* __restrict__ v3, const float* __restrict__ partials,
           int nb, const NegRec* __restrict__ neg, float* __restrict__ out) {
  __shared__ float red[TPB];
  float s = 0.0f;
  for (int i = threadIdx.x; i < nb; i += blockDim.x) s += partials[i];
  red[threadIdx.x] = s;
  __syncthreads();
  for (int st = TPB / 2; st > 0; st >>= 1) {
    if (threadIdx.x < st) red[threadIdx.x] += red[threadIdx.x + st];
    __syncthreads();
  }
  if (threadIdx.x == 0) {
    float total = -red[0];
    for (int j = 0; j < 22; ++j) {
      unsigned cnt = neg[j].count;
      if (cnt > 0u) {
        unsigned m1 = (unsigned)neg[j].mins;
        unsigned m2 = (unsigned)(neg[j].mins >> 32);
        int f1 = g_negf[j][0], f2 = g_negf[j][1], f3 = g_negf[j][2];
        // argmax(cumsum==f+1): (f+1)-th match if count>=f+1, else index 0
        unsigned i1 = (cnt >= (unsigned)(f1 + 1)) ? (f1 == 0 ? m1 : m2) : 0u;
        unsigned i2 = (cnt >= (unsigned)(f2 + 1)) ? (f2 == 0 ? m1 : m2) : 0u;
        unsigned i3 = (cnt >= (unsigned)(f3 + 1)) ? (f3 == 0 ? m1 : m2) : 0u;
        float t = 0.0f;
        for (int col = 0; col < 2; ++col)
          t += v1[2 * i1 + col] + v2[2 * i2 + col] - log1mexpf_dev(v3[2 * i3 + col]);
        total -= t;
      }
    }
    out[0] = total;
  }
}

extern "C" void kernel_launch(void* const* d_in, const int* in_sizes, int n_in,
                              void* d_out, int out_size, void* d_ws, size_t ws_size,
                              hipStream_t stream) {
  const float* v1 = (const float*)d_in[0];
  const float* v2 = (const float*)d_in[1];
  const float* v3 = (const float*)d_in[2];
  const int* xy   = (const int*)d_in[3];
  const int* yz   = (const int*)d_in[4];
  const int* xz   = (const int*)d_in[5];
  const int* flg  = (const int*)d_in[6];
  int B  = in_sizes[6];           // flag count == B
  int B4 = (B + 3) / 4;

  NegRec* neg     = (NegRec*)d_ws;
  float* partials = (float*)((char*)d_ws + 512);

  int blocks = (B4 + TPB - 1) / TPB;
  if (blocks > 4096) blocks = 4096;
  int maxPart = (int)((ws_size > 512 ? ws_size - 512 : 0) / sizeof(float));
  if (maxPart < 1) maxPart = 1;
  if (blocks > maxPart) blocks = maxPart;
  if (blocks < 1) blocks = 1;

  init_ws<<<1, 32, 0, stream>>>(neg);
  loss_main<<<blocks, TPB, 0, stream>>>(v1, v2, v3, xy, yz, xz, flg, B, B4,
                                        partials, neg);
  loss_final<<<1, TPB, 0, stream>>>(v1, v2, v3, partials, blocks, neg,
                                    (float*)d_out);
}